// LengthRegulator_1039382086246
// MI455X (gfx1250) — compile-verified
//
#include <hip/hip_runtime.h>

// LengthRegulator for MI455X (gfx1250, wave32).
//
// ys[b,t,:] = xs[b, i(t), :] where i(t) is the unique token whose repeat
// interval [left_i, cum_i) contains t, else 0.  Memory-bound gather:
//   writes 50.3 MB (NT stores), reads ~6.3 MB (xs fits in L2) -> ~2.4 us
//   at 23.3 TB/s.  A WMMA/GEMM formulation would be >= 5x slower.

#define BATCH 8
#define TIN   512
#define TOUT_ 4096
#define DIM   384   // 384 floats = 1536 B = 32 lanes * 3 * float4

typedef __attribute__((ext_vector_type(4))) float v4f;

// ---------------------------------------------------------------------------
// Kernel 1: per-batch inclusive scan of ds in LDS + build frame->token map.
// One 512-thread block per batch (16 wave32s).  map[b*TOUT_ + t] = token index
// covering frame t, or -1 past the total length.
// ---------------------------------------------------------------------------
__global__ __launch_bounds__(TIN) void lr_scan_map_kernel(
    const int* __restrict__ ds, int* __restrict__ map) {
  __shared__ int s[TIN];

  const int b = blockIdx.x;
  const int i = threadIdx.x;

  const int d = ds[b * TIN + i];   // 0..7
  s[i] = d;
  __syncthreads();

  // Hillis-Steele inclusive scan over 512 elements in LDS.
  #pragma unroll
  for (int off = 1; off < TIN; off <<= 1) {
    int v = (i >= off) ? s[i - off] : 0;
    __syncthreads();
    s[i] += v;
    __syncthreads();
  }

  const int cum  = s[i];
  const int left = cum - d;
  int* bmap = map + b * TOUT_;

  // Each token writes its (at most 7) covered frames.  Intervals are
  // disjoint, so no write conflicts.
  for (int r = 0; r < d; ++r) bmap[left + r] = i;

  // Tail frames (t >= total length) produce zeros.
  const int total = s[TIN - 1];    // max 512*7 = 3584 < 4096
  for (int t = total + i; t < TOUT_; t += TIN) bmap[t] = -1;
}

// ---------------------------------------------------------------------------
// Kernel 2: one wave32 per output row.  idx = map[row] is wave-uniform, so
// the branch is non-divergent.  Each lane moves 3 x float4 (b128 vmem ops);
// output is write-once -> non-temporal stores keep it out of L2.
// ---------------------------------------------------------------------------
__global__ __launch_bounds__(256) void lr_gather_kernel(
    const float* __restrict__ xs, const int* __restrict__ map,
    float* __restrict__ out) {
  const int wave = threadIdx.x >> 5;      // 8 waves per block
  const int lane = threadIdx.x & 31;
  const int row  = blockIdx.x * 8 + wave; // 0 .. BATCH*TOUT_-1
  const int b    = row >> 12;             // TOUT_ = 4096 rows per batch

  const int idx = map[row];               // wave-uniform (broadcast load)

  v4f* __restrict__ dst = (v4f*)(out + (size_t)row * DIM);

  if (idx >= 0) {
    const v4f* __restrict__ src =
        (const v4f*)(xs + ((size_t)b * TIN + (size_t)idx) * DIM);
    #pragma unroll
    for (int j = 0; j < 3; ++j) {
      v4f v = src[lane + 32 * j];                       // global_load_b128
      __builtin_nontemporal_store(v, dst + lane + 32 * j); // NT global_store_b128
    }
  } else {
    const v4f z = {0.0f, 0.0f, 0.0f, 0.0f};
    #pragma unroll
    for (int j = 0; j < 3; ++j)
      __builtin_nontemporal_store(z, dst + lane + 32 * j);
  }
}

// ---------------------------------------------------------------------------
extern "C" void kernel_launch(void* const* d_in, const int* in_sizes, int n_in,
                              void* d_out, int out_size, void* d_ws,
                              size_t ws_size, hipStream_t stream) {
  const float* xs = (const float*)d_in[0];   // (8, 512, 384) fp32
  const int*   ds = (const int*)d_in[1];     // (8, 512) int
  float*       out = (float*)d_out;          // (8, 4096, 384) fp32
  int*         map = (int*)d_ws;             // 8*4096*4 = 128 KB scratch

  lr_scan_map_kernel<<<BATCH, TIN, 0, stream>>>(ds, map);

  const int rows = BATCH * TOUT_;            // 32768 rows
  lr_gather_kernel<<<rows / 8, 256, 0, stream>>>(xs, map, out);
}